// NNPolicy_16956530884999
// MI455X (gfx1250) — compile-verified
//
#include <hip/hip_runtime.h>
#include <math.h>

typedef float v2f __attribute__((ext_vector_type(2)));
typedef float v8f __attribute__((ext_vector_type(8)));

#define TSTEPS   16
#define HDIM     64
#define ODIM     4
#define INDIM    8
#define LDS_STRIDE 68   // 64 + 4 pad: conflict-free row & column access

static __device__ __forceinline__ v8f wmma_f32_16x16x4(v2f a, v2f b, v8f c) {
  // 8 args: (neg_a, A, neg_b, B, c_mod, C, reuse_a, reuse_b)
  return __builtin_amdgcn_wmma_f32_16x16x4_f32(false, a, false, b, (short)0, c,
                                               false, false);
}

// 1-ulp hardware reciprocal (v_rcp_f32): avoids the ~9-instruction IEEE divide.
static __device__ __forceinline__ float fast_rcp(float x) {
  return __builtin_amdgcn_rcpf(x);
}

__global__ __launch_bounds__(256) void snn_fused_kernel(
    const float* __restrict__ x, const float* __restrict__ W1,
    const float* __restrict__ b1, const float* __restrict__ Wa,
    const float* __restrict__ ba, const float* __restrict__ tau_adp_h1,
    const float* __restrict__ tau_m_h1, const float* __restrict__ tau_m_a,
    const float* __restrict__ hid1_mem0, const float* __restrict__ a_out_mem0,
    float* __restrict__ out_val, float* __restrict__ out_act) {
  __shared__ float lds_spk[8][16 * LDS_STRIDE];  // per-wave transpose staging

  const int lane = threadIdx.x & 31;
  const int wave = threadIdx.x >> 5;
  const int tile = blockIdx.x * 8 + wave;  // 16 batch rows per wave
  const int row0 = tile * 16;
  const int nlo  = lane & 15;  // column index within 16-wide tile (C/D, B layouts)
  const int hi   = lane >> 4;  // lane half
  const int n4   = nlo & 3;    // clamped output-column index (always in-bounds)
  float* sspk = lds_spk[wave];

  // ---- per-lane per-H-tile constants (column h = t*16 + nlo) ----
  float alpha_h[4], ro_h[4], bias1[4];
#pragma unroll
  for (int t = 0; t < 4; ++t) {
    int h = t * 16 + nlo;
    alpha_h[t] = __expf(-fast_rcp(tau_m_h1[h]));
    ro_h[t]    = __expf(-fast_rcp(tau_adp_h1[h]));
    bias1[t]   = b1[h];
  }
  // Output-layer per-lane constants. Lanes with nlo>=4 duplicate column nlo&3:
  // their D columns / softmax results are never stored, so no masking needed.
  const float alpha_a = __expf(-fast_rcp(tau_m_a[n4]));
  const float oma     = 1.0f - alpha_a;
  const float cba     = oma * ba[n4];  // folded (1-alpha_a)*ba term

  // ---- Wa -> WMMA B-fragments (16 chunks of K=4 over H) ----
  // B layout (4x16 f32): VGPR r, lane L -> K = r + 2*(L>=16), N = L&15.
  // Unconditional clamped loads: no exec-mask branches; dup columns are dead.
  v2f wafrag[16];
#pragma unroll
  for (int c = 0; c < 16; ++c) {
    int k = c * 4 + 2 * hi;
    v2f f;
    f.x = Wa[n4 * HDIM + k];
    f.y = Wa[n4 * HDIM + k + 1];
    wafrag[c] = f;
  }

  // ---- W1^T -> B-fragments: tile t over H, chunk c over IN (K=8 -> 2 chunks) ----
  v2f w1frag[4][2];
#pragma unroll
  for (int t = 0; t < 4; ++t) {
#pragma unroll
    for (int c = 0; c < 2; ++c) {
      int h = t * 16 + nlo;
      int k = c * 4 + 2 * hi;
      v2f f;
      f.x = W1[h * INDIM + k];
      f.y = W1[h * INDIM + k + 1];
      w1frag[t][c] = f;
    }
  }

  // ---- x -> A-fragments. A layout (16x4 f32): lane L -> M = L&15, K = r + 2*(L>=16)
  const int am = nlo;  // A-matrix row held by this lane
  v2f xfrag[2];
#pragma unroll
  for (int c = 0; c < 2; ++c) {
    int k = c * 4 + 2 * hi;
    v2f f;
    f.x = x[(row0 + am) * INDIM + k];
    f.y = x[(row0 + am) * INDIM + k + 1];
    xfrag[c] = f;
  }

  // ---- inpt2 = (1-alpha_h) * (x @ W1^T + b1): pre-scaled drive (loop-invariant)
  float inpt2[4][8];
#pragma unroll
  for (int t = 0; t < 4; ++t) {
    v8f d = {};
    d = wmma_f32_16x16x4(xfrag[0], w1frag[t][0], d);
    d = wmma_f32_16x16x4(xfrag[1], w1frag[t][1], d);
    float omah = 1.0f - alpha_h[t];
#pragma unroll
    for (int r = 0; r < 8; ++r) inpt2[t][r] = omah * (d[r] + bias1[t]);
  }

  // ---- initial state in C-layout: row m = r + 8*hi, column h = t*16 + nlo ----
  float mem[4][8], bb[4][8], spk[4][8];
#pragma unroll
  for (int t = 0; t < 4; ++t) {
    int h = t * 16 + nlo;
#pragma unroll
    for (int r = 0; r < 8; ++r) {
      int row = row0 + r + 8 * hi;
      mem[t][r] = hid1_mem0[row * HDIM + h];
      bb[t][r]  = 0.01f;  // B_J0
      spk[t][r] = 0.0f;
    }
  }
  float a_mem[8], act[8];
#pragma unroll
  for (int r = 0; r < 8; ++r) {
    int row = row0 + r + 8 * hi;
    a_mem[r] = a_out_mem0[row * ODIM + n4];  // clamped: always in-bounds
    act[r]   = 0.0f;
  }

  // ---- recurrent time loop: all state stays in VGPRs ----
  for (int step = 0; step < TSTEPS; ++step) {
    // neuron update (elementwise in C-layout); stage new spikes into LDS
#pragma unroll
    for (int t = 0; t < 4; ++t) {
      float al = alpha_h[t], ro = ro_h[t], omr = 1.0f - ro_h[t];
#pragma unroll
      for (int r = 0; r < 8; ++r) {
        float s   = spk[t][r];
        float bn  = __builtin_fmaf(omr, s, ro * bb[t][r]);
        float Bth = __builtin_fmaf(1.8f, bn, 0.01f);
        float mn  = __builtin_fmaf(mem[t][r], al, inpt2[t][r]);
        mn        = __builtin_fmaf(-Bth, s, mn);
        float sn  = (mn - Bth) > 0.0f ? 1.0f : 0.0f;
        bb[t][r]  = bn;
        mem[t][r] = mn;
        spk[t][r] = sn;
        sspk[(r + 8 * hi) * LDS_STRIDE + t * 16 + nlo] = sn;  // row-major 16x64
      }
    }
    __builtin_amdgcn_wave_barrier();  // order per-wave LDS stores before loads

    // ia = spk @ Wa^T : K=64 as 16 chained K=4 WMMAs; reload spk in A-layout
    v8f d = {};
#pragma unroll
    for (int c = 0; c < 16; ++c) {
      int k = c * 4 + 2 * hi;
      v2f a = *(const v2f*)&sspk[am * LDS_STRIDE + k];  // 8B-aligned, bank-clean
      d = wmma_f32_16x16x4(a, wafrag[c], d);
    }
    __builtin_amdgcn_wave_barrier();  // protect LDS WAR vs next step's stores

    // a_mem update + softmax over O=4 (groups of 4 lanes) + accumulate act
#pragma unroll
    for (int r = 0; r < 8; ++r) {
      // a_mem = alpha_a*a_mem + (1-alpha_a)*(d + ba)  [ba folded into cba]
      float amem = __builtin_fmaf(a_mem[r], alpha_a,
                                  __builtin_fmaf(oma, d[r], cba));
      a_mem[r] = amem;
      float mx = amem;
      mx = fmaxf(mx, __shfl_xor(mx, 1, 32));
      mx = fmaxf(mx, __shfl_xor(mx, 2, 32));
      float e = __expf(amem - mx);
      float ssum = e;
      ssum += __shfl_xor(ssum, 1, 32);
      ssum += __shfl_xor(ssum, 2, 32);
      act[r] = __builtin_fmaf(e, fast_rcp(ssum), act[r]);  // rcp+fma, no IEEE div
    }
  }

  // ---- outputs ----
  // val = sum of softmax over a single logit per step == TIME_STEP, exactly.
  if (hi == 0) out_val[row0 + nlo] = (float)TSTEPS;
  if (nlo < ODIM) {
#pragma unroll
    for (int r = 0; r < 8; ++r) {
      int row = row0 + r + 8 * hi;
      out_act[row * ODIM + nlo] = act[r];
    }
  }
}

extern "C" void kernel_launch(void* const* d_in, const int* in_sizes, int n_in,
                              void* d_out, int out_size, void* d_ws,
                              size_t ws_size, hipStream_t stream) {
  const float* x          = (const float*)d_in[0];
  const float* W1         = (const float*)d_in[1];
  const float* b1         = (const float*)d_in[2];
  const float* Wa         = (const float*)d_in[3];
  const float* ba         = (const float*)d_in[4];
  // d_in[5]=Wc, d_in[6]=bc, d_in[10]=tau_m_c, d_in[13]=c_out_mem0 are dead:
  // softmax over the single-column c_mem is identically 1.0, so val == TIME_STEP.
  const float* tau_adp_h1 = (const float*)d_in[7];
  const float* tau_m_h1   = (const float*)d_in[8];
  const float* tau_m_a    = (const float*)d_in[9];
  const float* hid1_mem0  = (const float*)d_in[11];
  const float* a_out_mem0 = (const float*)d_in[12];

  const int B = in_sizes[0] / INDIM;  // 32768
  float* out_val = (float*)d_out;     // [B,1]
  float* out_act = out_val + B;       // [B,4]

  const int waves_total = B / 16;     // one wave per 16-row batch tile
  dim3 block(256);                    // 8 waves per block
  dim3 grid(waves_total / 8);
  snn_fused_kernel<<<grid, block, 0, stream>>>(x, W1, b1, Wa, ba, tau_adp_h1,
                                               tau_m_h1, tau_m_a, hid1_mem0,
                                               a_out_mem0, out_val, out_act);
}